// UNet_59717225284152
// MI455X (gfx1250) — compile-verified
//
#include <hip/hip_runtime.h>
#include <hip/hip_bf16.h>

// ---------------------------------------------------------------------------
// Sparse 3D U-Net for MI455X (gfx1250), fp32 end-to-end.
//
//   * Dense int32 index grids per level (96^3/48^3/24^3/12^3, ~4 MB total,
//     L2-resident) give O(1) neighbor lookup.
//   * Gather-GEMM-scatter conv: one wave32 owns a 16-point x 16-outchannel
//     tile mapped onto the 16x16 f32 C/D WMMA layout; per kernel offset it
//     chains CIN/4 V_WMMA_F32_16X16X4_F32 ops (fp32 matches reference
//     numerics; workload is L2-gather bound, not math bound).
//   * 4 waves / workgroup. Weights staged into LDS in 9-offset chunks with
//     float4 copies (global_load_b128 -> ds_store_b128); B feeds the WMMAs
//     from conflict-free ds_load_2addr_b32 pairs. 3 barrier pairs per sm-conv
//     instead of 27.
//   * COUT is a template parameter: all strides are immediates; full tiles
//     take a uniform store path (8x global_store_b32, no exec juggling).
//   * Missing neighbors / tail points: A-row pointer cndmask'd to a zeroed
//     row -- no gating multiplies, EXEC stays all-ones through WMMA.
// ---------------------------------------------------------------------------

typedef float v2f __attribute__((ext_vector_type(2)));
typedef float v8f __attribute__((ext_vector_type(8)));

__device__ __forceinline__ v8f wmma_f32_k4(v2f a, v2f b, v8f c) {
  // 8 args: (neg_a, A, neg_b, B, c_mod, C, reuse_a, reuse_b)
  return __builtin_amdgcn_wmma_f32_16x16x4_f32(false, a, false, b, (short)0, c,
                                               false, false);
}

// ---------------------------------------------------------------------------
// Utility kernels
// ---------------------------------------------------------------------------
__global__ void fill_i32_kernel(int* __restrict__ p, int n, int v) {
  int i = blockIdx.x * blockDim.x + threadIdx.x;
  if (i < n) p[i] = v;
}

__global__ void scatter_idx_kernel(int* __restrict__ grid,
                                   const int* __restrict__ pos, int N, int g) {
  int i = blockIdx.x * blockDim.x + threadIdx.x;
  if (i < N) {
    int x = pos[i * 3 + 0], y = pos[i * 3 + 1], z = pos[i * 3 + 2];
    grid[(x * g + y) * g + z] = i;
  }
}

// One block per point, blockDim.x == C (channel id = threadIdx.x, no divides).
__global__ void bnrelu_kernel(const float* __restrict__ x, float* __restrict__ y,
                              const float* __restrict__ gam,
                              const float* __restrict__ bet,
                              const float* __restrict__ mean,
                              const float* __restrict__ var) {
  const int c = threadIdx.x;
  const int i = blockIdx.x * blockDim.x + c;
  float v = gam[c] * (x[i] - mean[c]) * rsqrtf(var[c] + 1e-4f) + bet[c];
  y[i] = v > 0.f ? v : 0.f;
}

// One block per point, blockDim.x == Ca + Cb.
__global__ void concat_kernel(const float* __restrict__ a,
                              const float* __restrict__ b, float* __restrict__ y,
                              int Ca) {
  const int c = threadIdx.x;
  const int C = blockDim.x;
  const int r = blockIdx.x;
  y[(size_t)r * C + c] =
      (c < Ca) ? a[(size_t)r * Ca + c] : b[(size_t)r * (C - Ca) + (c - Ca)];
}

// ---------------------------------------------------------------------------
// Shared inner GEMM fragment: A gathered from global (or zero row), B from
// LDS slice bbase (layout (ci*16+co), conflict-free).
// ---------------------------------------------------------------------------
template <int CIN>
__device__ __forceinline__ void gemm_frag(const float* __restrict__ arow,
                                          const float* __restrict__ bbase,
                                          v8f& acc) {
#pragma unroll
  for (int k = 0; k < CIN; k += 4) {
    v2f a = *(const v2f*)(arow + k);
    v2f b;
    b.x = bbase[(k + 0) * 16];
    b.y = bbase[(k + 1) * 16];
    acc = wmma_f32_k4(a, b, acc);
  }
}

template <int COUT>
__device__ __forceinline__ void store_tile(float* __restrict__ fout, int m0,
                                           int cob, int row, int hi, int N,
                                           const v8f& acc) {
  if (m0 + 16 <= N) {  // wave-uniform fast path: constant-offset stores
    float* __restrict__ dst = fout + (size_t)(m0 + hi * 8) * COUT + cob + row;
#pragma unroll
    for (int r = 0; r < 8; ++r) dst[r * COUT] = acc[r];
  } else {
#pragma unroll
    for (int r = 0; r < 8; ++r) {
      const int m = m0 + r + hi * 8;
      if (m < N) fout[(size_t)m * COUT + cob + row] = acc[r];
    }
  }
}

// ---------------------------------------------------------------------------
// Submanifold 3x3x3 conv, stride 1, SAME, same in/out positions.
// Block: 128 threads = 4 waves = 4 consecutive 16-point M tiles, one 16-wide
// out-channel tile (blockIdx.y). Weights staged in 9-offset LDS chunks.
// Weight layout (kd,kh,kw,cin,cout): o = ((dx+1)*3+(dy+1))*3+(dz+1).
// ---------------------------------------------------------------------------
template <int CIN, int COUT>
__global__ __launch_bounds__(128) void sm_conv3_kernel(
    const float* __restrict__ fin, const int* __restrict__ pos,
    const int* __restrict__ grid, int g, int N, const float* __restrict__ W,
    const float* __restrict__ zrow, float* __restrict__ fout) {
  __shared__ float Bs[9 * CIN * 16];
  const int tid = threadIdx.x;
  const int lane = tid & 31;
  const int wv = tid >> 5;
  const int row = lane & 15;
  const int hi = lane >> 4;  // 0: K=k0..k0+1, 1: K=k0+2..k0+3 (ISA A layout)
  const int m0 = (blockIdx.x * 4 + wv) << 4;
  const int cob = blockIdx.y << 4;

  const int p = m0 + row;
  const bool pv = p < N;
  const int pc = pv ? p : (N - 1);
  const int px = pos[pc * 3 + 0], py = pos[pc * 3 + 1], pz = pos[pc * 3 + 2];

  v8f acc = {};
#pragma unroll 1
  for (int og = 0; og < 27; og += 9) {
    __syncthreads();  // protect Bs reads of previous chunk
    // Stage rows og*CIN .. og*CIN + 9*CIN of the (cob..cob+15) column slice.
#pragma unroll 1
    for (int idx = tid; idx < 9 * CIN * 4; idx += 128) {
      const int r = idx >> 2, c4 = idx & 3;
      *(float4*)&Bs[r * 16 + c4 * 4] =
          *(const float4*)&W[(size_t)(og * CIN + r) * COUT + cob + c4 * 4];
    }
    __syncthreads();

#pragma unroll 1
    for (int oi = 0; oi < 9; ++oi) {
      const int o = og + oi;
      // Branchless neighbor lookup (keep EXEC stable).
      const int dx = o / 9 - 1, dy = (o / 3) % 3 - 1, dz = o % 3 - 1;
      const int nx = px + dx, ny = py + dy, nz = pz + dz;
      const bool inb = ((unsigned)nx < (unsigned)g) &
                       ((unsigned)ny < (unsigned)g) &
                       ((unsigned)nz < (unsigned)g);
      const int nxc = inb ? nx : 0, nyc = inb ? ny : 0, nzc = inb ? nz : 0;
      int nbr = grid[(nxc * g + nyc) * g + nzc];
      nbr = (inb && pv) ? nbr : -1;

      const float* __restrict__ arow =
          ((nbr < 0) ? zrow : (fin + (size_t)nbr * CIN)) + hi * 2;
      const float* __restrict__ bbase = Bs + (oi * CIN + hi * 2) * 16 + row;
      gemm_frag<CIN>(arow, bbase, acc);
    }
  }
  store_tile<COUT>(fout, m0, cob, row, hi, N, acc);
}

// ---------------------------------------------------------------------------
// Down conv: 2x2x2, stride 2, VALID. Output at coarse pos q gathers fine
// positions 2q+d, d in {0,1}^3 (always in range). o = (dx*2+dy)*2+dz.
// All 8 offsets staged once.
// ---------------------------------------------------------------------------
template <int CIN, int COUT>
__global__ __launch_bounds__(128) void down_conv_kernel(
    const float* __restrict__ fin, const int* __restrict__ posc,
    const int* __restrict__ gridf, int gf, int Nc, const float* __restrict__ W,
    const float* __restrict__ zrow, float* __restrict__ fout) {
  __shared__ float Bs[8 * CIN * 16];
  const int tid = threadIdx.x;
  const int lane = tid & 31;
  const int wv = tid >> 5;
  const int row = lane & 15;
  const int hi = lane >> 4;
  const int m0 = (blockIdx.x * 4 + wv) << 4;
  const int cob = blockIdx.y << 4;

  const int p = m0 + row;
  const bool pv = p < Nc;
  const int pc = pv ? p : (Nc - 1);
  const int px = posc[pc * 3 + 0], py = posc[pc * 3 + 1], pz = posc[pc * 3 + 2];

#pragma unroll 1
  for (int idx = tid; idx < 8 * CIN * 4; idx += 128) {
    const int r = idx >> 2, c4 = idx & 3;
    *(float4*)&Bs[r * 16 + c4 * 4] =
        *(const float4*)&W[(size_t)r * COUT + cob + c4 * 4];
  }
  __syncthreads();

  v8f acc = {};
#pragma unroll 1
  for (int o = 0; o < 8; ++o) {
    const int dx = (o >> 2) & 1, dy = (o >> 1) & 1, dz = o & 1;
    const int nx = 2 * px + dx, ny = 2 * py + dy, nz = 2 * pz + dz;
    int nbr = gridf[(nx * gf + ny) * gf + nz];
    nbr = pv ? nbr : -1;

    const float* __restrict__ arow =
        ((nbr < 0) ? zrow : (fin + (size_t)nbr * CIN)) + hi * 2;
    const float* __restrict__ bbase = Bs + (o * CIN + hi * 2) * 16 + row;
    gemm_frag<CIN>(arow, bbase, acc);
  }
  store_tile<COUT>(fout, m0, cob, row, hi, Nc, acc);
}

// ---------------------------------------------------------------------------
// Up (transpose) conv: 2x2x2, stride 2, VALID. Each fine point p has exactly
// one contributing coarse parent q = p>>1 with kernel offset k = p&1 per dim.
// Implemented as 8 masked offsets so B stays uniform per WMMA.
// ---------------------------------------------------------------------------
template <int CIN, int COUT>
__global__ __launch_bounds__(128) void up_conv_kernel(
    const float* __restrict__ fin /*coarse feats*/, const int* __restrict__ posf,
    const int* __restrict__ gridc, int gc, int Nf, const float* __restrict__ W,
    const float* __restrict__ zrow, float* __restrict__ fout) {
  __shared__ float Bs[8 * CIN * 16];
  const int tid = threadIdx.x;
  const int lane = tid & 31;
  const int wv = tid >> 5;
  const int row = lane & 15;
  const int hi = lane >> 4;
  const int m0 = (blockIdx.x * 4 + wv) << 4;
  const int cob = blockIdx.y << 4;

  const int p = m0 + row;
  const bool pv = p < Nf;
  const int pc = pv ? p : (Nf - 1);
  const int px = posf[pc * 3 + 0], py = posf[pc * 3 + 1], pz = posf[pc * 3 + 2];
  const int ko = ((px & 1) * 2 + (py & 1)) * 2 + (pz & 1);
  const int q = gridc[(((px >> 1) * gc) + (py >> 1)) * gc + (pz >> 1)];

#pragma unroll 1
  for (int idx = tid; idx < 8 * CIN * 4; idx += 128) {
    const int r = idx >> 2, c4 = idx & 3;
    *(float4*)&Bs[r * 16 + c4 * 4] =
        *(const float4*)&W[(size_t)r * COUT + cob + c4 * 4];
  }
  __syncthreads();

  v8f acc = {};
#pragma unroll 1
  for (int o = 0; o < 8; ++o) {
    const int nbr = (pv && (ko == o)) ? q : -1;
    const float* __restrict__ arow =
        ((nbr < 0) ? zrow : (fin + (size_t)nbr * CIN)) + hi * 2;
    const float* __restrict__ bbase = Bs + (o * CIN + hi * 2) * 16 + row;
    gemm_frag<CIN>(arow, bbase, acc);
  }
  store_tile<COUT>(fout, m0, cob, row, hi, Nf, acc);
}

// ---------------------------------------------------------------------------
// Host side
// ---------------------------------------------------------------------------
// Input flattening assumption (pytree-style, dicts traversed in sorted key
// order): 0=feat; 1..65=params sorted alphabetically; 66..69=pos0..pos3.
//   bn groups hold (_b,_g,_m,_v) in that (sorted) order.
static inline int bn1_idx(int l) { return 1 + 4 * l; }    // l = 0..3
static inline int bn2_idx(int l) { return 17 + 4 * l; }   // l = 0..2
static inline int bn3_idx(int l) { return 29 + 4 * l; }   // l = 0..2
static inline int bnup_idx(int l) { return 41 + 4 * l; }  // l = 0..2
static inline int w_down_idx(int l) { return 53 + l; }
static inline int w_sm1_idx(int l) { return 56 + l; }
static inline int w_sm2_idx(int l) { return 60 + l; }
static inline int w_up_idx(int l) { return 63 + l; }

static void launch_bnrelu(const float* x, float* y, void* const* d_in, int base,
                          int N, int C, hipStream_t s) {
  const float* b = (const float*)d_in[base + 0];
  const float* g = (const float*)d_in[base + 1];
  const float* m = (const float*)d_in[base + 2];
  const float* v = (const float*)d_in[base + 3];
  bnrelu_kernel<<<N, C, 0, s>>>(x, y, g, b, m, v);
}

static void launch_sm(const float* fin, const int* pos, const int* grid, int g,
                      int N, const float* W, int Cin, int Cout,
                      const float* zrow, float* fout, hipStream_t s) {
  dim3 gr((N + 63) / 64, Cout / 16), bl(128);
#define SM_CASE(ci, co)                                                       \
  if (Cin == ci && Cout == co) {                                              \
    sm_conv3_kernel<ci, co>                                                   \
        <<<gr, bl, 0, s>>>(fin, pos, grid, g, N, W, zrow, fout);              \
    return;                                                                   \
  }
  SM_CASE(16, 16)   // sm1 level 0
  SM_CASE(32, 32)   // sm1 level 1
  SM_CASE(48, 48)   // sm1 level 2
  SM_CASE(64, 64)   // sm1 level 3
  SM_CASE(32, 16)   // sm2 level 0
  SM_CASE(64, 32)   // sm2 level 1
  SM_CASE(96, 48)   // sm2 level 2
#undef SM_CASE
}

static void launch_down(const float* fin, const int* posc, const int* gridf,
                        int gf, int Nc, const float* W, int Cin, int Cout,
                        const float* zrow, float* fout, hipStream_t s) {
  dim3 gr((Nc + 63) / 64, Cout / 16), bl(128);
#define DN_CASE(ci, co)                                                       \
  if (Cin == ci && Cout == co) {                                              \
    down_conv_kernel<ci, co>                                                  \
        <<<gr, bl, 0, s>>>(fin, posc, gridf, gf, Nc, W, zrow, fout);          \
    return;                                                                   \
  }
  DN_CASE(16, 32)
  DN_CASE(32, 48)
  DN_CASE(48, 64)
#undef DN_CASE
}

static void launch_up(const float* fin, const int* posf, const int* gridc,
                      int gc, int Nf, const float* W, int Cin, int Cout,
                      const float* zrow, float* fout, hipStream_t s) {
  dim3 gr((Nf + 63) / 64, Cout / 16), bl(128);
#define UP_CASE(ci, co)                                                       \
  if (Cin == ci && Cout == co) {                                              \
    up_conv_kernel<ci, co>                                                    \
        <<<gr, bl, 0, s>>>(fin, posf, gridc, gc, Nf, W, zrow, fout);          \
    return;                                                                   \
  }
  UP_CASE(32, 16)
  UP_CASE(48, 32)
  UP_CASE(64, 48)
#undef UP_CASE
}

extern "C" void kernel_launch(void* const* d_in, const int* in_sizes, int n_in,
                              void* d_out, int out_size, void* d_ws,
                              size_t ws_size, hipStream_t stream) {
  (void)n_in; (void)out_size; (void)ws_size;
  static const int G_[4] = {96, 48, 24, 12};
  static const int PL[4] = {16, 32, 48, 64};

  const float* feat = (const float*)d_in[0];
  const int* pos[4];
  int N[4];
  for (int l = 0; l < 4; ++l) {
    pos[l] = (const int*)d_in[66 + l];
    N[l] = in_sizes[66 + l] / 3;
  }

  // Bump allocator over d_ws (poisoned 0xAA by harness -> we init everything).
  char* wsb = (char*)d_ws;
  size_t off = 0;
  auto alloc = [&](size_t bytes) -> void* {
    void* p = wsb + off;
    off += (bytes + 255) & ~(size_t)255;
    return p;
  };

  int* grid[4];
  for (int l = 0; l < 4; ++l)
    grid[l] = (int*)alloc((size_t)G_[l] * G_[l] * G_[l] * sizeof(int));
  float* zrow = (float*)alloc(128 * sizeof(float));  // zeroed A row
  float* skip0 = (float*)alloc((size_t)N[0] * 16 * sizeof(float));
  float* skip1 = (float*)alloc((size_t)N[1] * 32 * sizeof(float));
  float* skip2 = (float*)alloc((size_t)N[2] * 48 * sizeof(float));
  size_t tmax = 0;
  {
    size_t c[4] = {(size_t)N[0] * 32, (size_t)N[1] * 64, (size_t)N[2] * 96,
                   (size_t)N[3] * 64};
    for (int i = 0; i < 4; ++i) tmax = c[i] > tmax ? c[i] : tmax;
  }
  float* T0 = (float*)alloc(tmax * sizeof(float));
  float* T1 = (float*)alloc(tmax * sizeof(float));
  float* T2 = (float*)alloc(tmax * sizeof(float));
  float* outf = (float*)d_out;

  // --- Init zero row + per-level index grids ---
  fill_i32_kernel<<<1, 128, 0, stream>>>((int*)zrow, 128, 0);
  for (int l = 0; l < 4; ++l) {
    int cells = G_[l] * G_[l] * G_[l];
    fill_i32_kernel<<<(cells + 255) / 256, 256, 0, stream>>>(grid[l], cells, -1);
    scatter_idx_kernel<<<(N[l] + 255) / 256, 256, 0, stream>>>(grid[l], pos[l],
                                                               N[l], G_[l]);
  }

  const float* skips[3] = {skip0, skip1, skip2};

  // --- Encoder: levels 0..2  (bn1+relu -> sm1 -> skip -> bn2+relu -> down) ---
  const float* cur = feat;  // input features at level l (N[l] x PL[l])
  for (int l = 0; l < 3; ++l) {
    int a = PL[l], b = PL[l + 1];
    float* skip = (float*)skips[l];
    launch_bnrelu(cur, T0, d_in, bn1_idx(l), N[l], a, stream);
    launch_sm(T0, pos[l], grid[l], G_[l], N[l],
              (const float*)d_in[w_sm1_idx(l)], a, a, zrow, skip, stream);
    launch_bnrelu(skip, T0, d_in, bn2_idx(l), N[l], a, stream);
    launch_down(T0, pos[l + 1], grid[l], G_[l], N[l + 1],
                (const float*)d_in[w_down_idx(l)], a, b, zrow, T1, stream);
    cur = T1;
  }

  // --- Bottleneck: level 3 (bn1+relu -> sm1) ---
  launch_bnrelu(cur, T0, d_in, bn1_idx(3), N[3], 64, stream);
  launch_sm(T0, pos[3], grid[3], G_[3], N[3], (const float*)d_in[w_sm1_idx(3)],
            64, 64, zrow, T1, stream);

  // --- Decoder: levels 2..0 ---
  // h(=T1, channels b at level l+1): bn_up+relu -> up conv -> concat(skip,.)
  // -> bn3+relu -> sm2  (output channels a; final level writes d_out)
  for (int l = 2; l >= 0; --l) {
    int a = PL[l], b = PL[l + 1];
    launch_bnrelu(T1, T0, d_in, bnup_idx(l), N[l + 1], b, stream);
    launch_up(T0, pos[l], grid[l + 1], G_[l + 1], N[l],
              (const float*)d_in[w_up_idx(l)], b, a, zrow, T1, stream);
    concat_kernel<<<N[l], 2 * a, 0, stream>>>(skips[l], T1, T2, a);
    launch_bnrelu(T2, T0, d_in, bn3_idx(l), N[l], 2 * a, stream);
    float* dst = (l == 0) ? outf : T1;
    launch_sm(T0, pos[l], grid[l], G_[l], N[l],
              (const float*)d_in[w_sm2_idx(l)], 2 * a, a, zrow, dst, stream);
  }
}